// ManifoldSeparationLoss_20126216749257
// MI455X (gfx1250) — compile-verified
//
#include <hip/hip_runtime.h>
#include <hip/hip_bf16.h>
#include <math.h>

typedef __attribute__((ext_vector_type(2))) float v2f;
typedef __attribute__((ext_vector_type(8))) float v8f;

#define NUM_CLUSTERS 64
#define TAB (NUM_CLUSTERS * NUM_CLUSTERS)
#define INF_BITS 0x7F800000u
#define MARGIN 1.0f

// ---------------------------------------------------------------------------
// Kernel 1: init global 64x64 min table to +inf (as u32 bits)
// ---------------------------------------------------------------------------
__global__ void msl_init_ws(unsigned* __restrict__ ws) {
  int i = blockIdx.x * blockDim.x + threadIdx.x;
  if (i < TAB) ws[i] = INF_BITS;
}

// ---------------------------------------------------------------------------
// Kernel 2: V_WMMA_F32_16X16X4_F32 produces squared distances DIRECTLY:
//   A row m = ( x_m,   y_m,  |p_m|^2, 1      )
//   B col n = (-2x_n, -2y_n, 1,       |p_n|^2)
//   D[m,n]  = |p_m|^2 + |p_n|^2 - 2 p_m.p_n  = d^2(m,n)
// Lanes 0-15 carry K=0,1 fragments; lanes 16-31 carry K=2,3 fragments.
// Segmented min into 64x64 LDS table via ds_min_u32 (nonneg float bits are
// u32-monotone); entries with d^2 >= MARGIN^2 pruned (can't affect hinge).
// Only ti<=tj tiles computed (symmetry); finalize merges both directions.
// ---------------------------------------------------------------------------
__global__ void __launch_bounds__(256)
msl_pair_min(const float2* __restrict__ coords,
             const int* __restrict__ labels,
             unsigned* __restrict__ ws,
             int B) {
  __shared__ unsigned tab[TAB];   // 16 KB of the 320 KB WGP LDS
  const int tid = threadIdx.x;
  for (int i = tid; i < TAB; i += blockDim.x) tab[i] = INF_BITS;
  __syncthreads();

  const int lane = tid & 31;
  const int wavesPerBlock = blockDim.x >> 5;
  const int gwave = blockIdx.x * wavesPerBlock + (tid >> 5);
  const int nwaves = gridDim.x * wavesPerBlock;

  const int NT = B >> 4;                                  // 512 tiles per dim
  const long long T = (long long)NT * (NT + 1) / 2;       // upper-tri tiles

  const int r  = lane & 15;   // point slot in tile (lanes 16-31 duplicate)
  const int hi = lane >> 4;   // 1 for lanes 16-31 (K=2,3 half)

  for (long long t = gwave; t < T; t += nwaves) {
    // decode flat upper-triangular index -> (ti, tj) with ti <= tj
    int tj = (int)((sqrtf(8.0f * (float)t + 1.0f) - 1.0f) * 0.5f);
    while ((long long)(tj + 1) * (tj + 2) / 2 <= t) ++tj;
    while ((long long)tj * (tj + 1) / 2 > t) --tj;
    const int ti = (int)(t - (long long)tj * (tj + 1) / 2);

    const int ib = ti << 4, jb = tj << 4;
    const float2 ci = coords[ib + r];
    const float2 cj = coords[jb + r];
    const int    lj = labels[jb + r];          // this lane's column label

    // row labels for this lane's 8 output rows m = v + 8*hi (WGP$-resident,
    // batched under LOADcnt -> no DScnt serialization with the atomics)
    int lim[8];
#pragma unroll
    for (int v = 0; v < 8; ++v) lim[v] = labels[ib + v + (hi << 3)];

    const float sqi = ci.x * ci.x + ci.y * ci.y;
    const float sqj = cj.x * cj.x + cj.y * cj.y;

    v2f a, b;
    a[0] = hi ? sqi  : ci.x;           // K2 = |p_i|^2   | K0 = x_i
    a[1] = hi ? 1.0f : ci.y;           // K3 = 1         | K1 = y_i
    b[0] = hi ? 1.0f : -2.0f * cj.x;   // K2 = 1         | K0 = -2 x_j
    b[1] = hi ? sqj  : -2.0f * cj.y;   // K3 = |p_j|^2   | K1 = -2 y_j

    v8f acc = {};
    acc = __builtin_amdgcn_wmma_f32_16x16x4_f32(
        /*neg_a=*/false, a, /*neg_b=*/false, b,
        /*c_mod=*/(short)0, acc, /*reuse_a=*/false, /*reuse_b=*/false);

    // D layout: VGPR v, lanes 0-15 -> M=v, lanes 16-31 -> M=v+8; N = lane&15
    const int gj = jb + r;
#pragma unroll
    for (int v = 0; v < 8; ++v) {
      const int gi = ib + v + (hi << 3);
      float d2 = fmaxf(acc[v], 0.0f);
      if (gi == gj) d2 = 2.0f;            // self-pair: >= margin^2, pruned
      // d2 >= MARGIN^2 can never lower the hinge -> skip
      if (d2 < MARGIN * MARGIN) {
        atomicMin(&tab[lim[v] * NUM_CLUSTERS + lj], __float_as_uint(d2));
      }
    }
  }

  __syncthreads();
  for (int i = tid; i < TAB; i += blockDim.x) {
    const unsigned v = tab[i];
    if (v != INF_BITS) atomicMin(&ws[i], v);
  }
}

// ---------------------------------------------------------------------------
// Kernel 3: hinge + mean over the 2016 upper-triangle cluster pairs.
// Table is "directional" (row tile <= col tile), so true M = min(ws[a,b],
// ws[b,a]). Untouched entries stay +inf -> hinge contribution 0.
// ---------------------------------------------------------------------------
__global__ void msl_finalize(const unsigned* __restrict__ ws,
                             float* __restrict__ out) {
  __shared__ float red[256];
  float acc = 0.0f;
  for (int idx = threadIdx.x; idx < TAB; idx += 256) {
    const int aC = idx >> 6, bC = idx & 63;
    if (aC < bC) {
      const unsigned u0 = ws[aC * 64 + bC];
      const unsigned u1 = ws[bC * 64 + aC];
      const unsigned u  = u0 < u1 ? u0 : u1;
      const float m = sqrtf(__uint_as_float(u));
      acc += fmaxf(0.0f, MARGIN - m);
    }
  }
  red[threadIdx.x] = acc;
  __syncthreads();
  for (int s = 128; s > 0; s >>= 1) {
    if (threadIdx.x < s) red[threadIdx.x] += red[threadIdx.x + s];
    __syncthreads();
  }
  if (threadIdx.x == 0) {
    const float np = (float)(NUM_CLUSTERS * (NUM_CLUSTERS - 1) / 2);
    out[0] = red[0] / np;
  }
}

// ---------------------------------------------------------------------------
extern "C" void kernel_launch(void* const* d_in, const int* in_sizes, int n_in,
                              void* d_out, int out_size, void* d_ws, size_t ws_size,
                              hipStream_t stream) {
  // d_in[0] = features (unused), d_in[1] = cluster_labels (int32, B),
  // d_in[2] = manifold_coords (float32, Bx2)
  const int*    labels = (const int*)d_in[1];
  const float2* coords = (const float2*)d_in[2];
  unsigned*     ws     = (unsigned*)d_ws;       // 4096 * 4 B = 16 KB scratch
  const int B = in_sizes[1];

  msl_init_ws<<<(TAB + 255) / 256, 256, 0, stream>>>(ws);
  msl_pair_min<<<512, 256, 0, stream>>>(coords, labels, ws, B);
  msl_finalize<<<1, 256, 0, stream>>>(ws, (float*)d_out);
}